// DescriptorModule_11854109737448
// MI455X (gfx1250) — compile-verified
//
#include <hip/hip_runtime.h>
#include <hip/hip_bf16.h>
#include <math.h>

typedef float v2f __attribute__((ext_vector_type(2)));
typedef float v8f __attribute__((ext_vector_type(8)));

#define NSNAP 8
#define NPTS  4096
#define NN    48
#define NN0   24
#define EMBD  32
#define SUBD  16
#define BOXL  12.0f
#define RCS   2.0f
#define RCUT  3.0f
#define PAIR_STRIDE 744   // floats per pair-type weight pack
#define WTOTAL (3*PAIR_STRIDE)

// per-pair packed weight offsets (floats):
// L0 W@0(1x2)  b@2(2)   L1 W@4(2x4)   b@12(4)   L2 W@16(4x8) b@48(8)
// L3 W@56(8x16) b@184(16)  L4 W@200(16x32) b@712(32)

__global__ __launch_bounds__(128)
void descriptor_wmma_kernel(const float* __restrict__ pos,
                            const int*   __restrict__ types,
                            const int*   __restrict__ neigh,
                            const float* __restrict__ wts,
                            float*       __restrict__ out)
{
    __shared__ float sW[WTOTAL];          // packed MLP weights, all 3 pair types
    __shared__ float sS [4][NN];          // s_ij
    __shared__ float sRh[4][NN][3];       // r_hat (already scaled by s_ij*inv_r)
    __shared__ float sG [4][NN][EMBD];    // MLP output G
    __shared__ float sRm[4][4][SUBD];     // R  (4x16)
    __shared__ float sR1[4][NN][SUBD];    // R1 (48x16)

    const int tid  = threadIdx.x;
    const int w    = tid >> 5;            // wave id in block (0..3)
    const int lane = tid & 31;
    const int bp   = blockIdx.x * 4 + w;  // 0..32767 (b*4096 + p)
    const int b    = bp >> 12;

    // ---- cooperative weight load into LDS ----
    for (int i = tid; i < WTOTAL; i += 128) sW[i] = wts[i];
    __syncthreads();

    // ---- geometry + per-neighbor MLP (lane = neighbor; 2 passes) ----
    const float pix = pos[bp * 3 + 0];
    const float piy = pos[bp * 3 + 1];
    const float piz = pos[bp * 3 + 2];
    const int   a_t = types[bp];

    for (int t = 0; t < 2; ++t) {
        const int n = lane + 32 * t;
        if (n < NN) {
            const int j = neigh[(size_t)bp * NN + n];
            float dx = pos[(b * NPTS + j) * 3 + 0] - pix;
            float dy = pos[(b * NPTS + j) * 3 + 1] - piy;
            float dz = pos[(b * NPTS + j) * 3 + 2] - piz;
            dx -= BOXL * rintf(dx * (1.0f / BOXL));
            dy -= BOXL * rintf(dy * (1.0f / BOXL));
            dz -= BOXL * rintf(dz * (1.0f / BOXL));
            const float r = sqrtf(dx * dx + dy * dy + dz * dz);
            const float inv_r = (r > 1e-6f) ? (1.0f / fmaxf(r, 1e-6f)) : 0.0f;
            float sw;
            if (r < RCS)      sw = 1.0f;
            else if (r < RCUT) {
                const float u = (r - RCS) / (RCUT - RCS);
                sw = 0.5f * cosf(3.14159265358979323846f * u) + 0.5f;
            } else            sw = 0.0f;
            const float s = inv_r * sw;
            const float f = s * inv_r;
            sS [w][n]    = s;
            sRh[w][n][0] = dx * f;
            sRh[w][n][1] = dy * f;
            sRh[w][n][2] = dz * f;

            // ---- MLP: scalar s -> 32, pair = center_type + group ----
            const int pair = a_t + ((n >= NN0) ? 1 : 0);
            const float* Wp = &sW[pair * PAIR_STRIDE];
            float h0[2], h1[4], h2[8], h3[16];
            #pragma unroll
            for (int jj = 0; jj < 2; ++jj)
                h0[jj] = fmaxf(s * Wp[0 + jj] + Wp[2 + jj], 0.0f);
            #pragma unroll
            for (int jj = 0; jj < 4; ++jj) {
                float a = Wp[12 + jj];
                #pragma unroll
                for (int ii = 0; ii < 2; ++ii) a += h0[ii] * Wp[4 + ii * 4 + jj];
                h1[jj] = fmaxf(a, 0.0f);
            }
            #pragma unroll
            for (int jj = 0; jj < 8; ++jj) {
                float a = Wp[48 + jj];
                #pragma unroll
                for (int ii = 0; ii < 4; ++ii) a += h1[ii] * Wp[16 + ii * 8 + jj];
                h2[jj] = fmaxf(a, 0.0f);
            }
            #pragma unroll
            for (int jj = 0; jj < 16; ++jj) {
                float a = Wp[184 + jj];
                #pragma unroll
                for (int ii = 0; ii < 8; ++ii) a += h2[ii] * Wp[56 + ii * 16 + jj];
                h3[jj] = fmaxf(a, 0.0f);
            }
            #pragma unroll
            for (int jj = 0; jj < 32; ++jj) {
                float a = Wp[712 + jj];
                #pragma unroll
                for (int ii = 0; ii < 16; ++ii) a += h3[ii] * Wp[200 + ii * 32 + jj];
                sG[w][n][jj] = fmaxf(a, 0.0f);
            }
        }
    }
    __syncthreads();

    // ---- R[d][e] = sum_n r_tilde[n][d] * G[n][e<16]  (64 outputs, VALU) ----
    #pragma unroll
    for (int t = 0; t < 2; ++t) {
        const int entry = lane + 32 * t;     // 0..63
        const int d = entry >> 4;            // 0..3
        const int e = entry & 15;
        float acc = 0.0f;
        #pragma unroll 4
        for (int n = 0; n < NN; ++n) {
            const float rt = (d == 0) ? sS[w][n] : sRh[w][n][d - 1];
            acc += rt * sG[w][n][e];
        }
        sRm[w][d][e] = acc;
    }
    __syncthreads();

    // ---- WMMA fragment coordinates (wave32, f32 16x16x4 layouts) ----
    const int half = lane >> 4;     // 0: lanes 0-15, 1: lanes 16-31
    const int lm   = lane & 15;
    const int K0   = half * 2;      // K index of VGPR0 element

    // ---- R1 = r_tilde(48x4) x R(4x16): 3 WMMA tiles ----
    v2f Bfrag;
    Bfrag[0] = sRm[w][K0 + 0][lm];
    Bfrag[1] = sRm[w][K0 + 1][lm];
    #pragma unroll
    for (int tile = 0; tile < 3; ++tile) {
        const int m0 = tile * 16;
        const int n  = m0 + lm;
        v2f Afrag;
        // A[m][k] = r_tilde[n][K0+k]; r_tilde = [s, rh0, rh1, rh2]
        Afrag[0] = (K0 == 0) ? sS[w][n] : sRh[w][n][1];
        Afrag[1] = (K0 == 0) ? sRh[w][n][0] : sRh[w][n][2];
        v8f C = {};
        C = __builtin_amdgcn_wmma_f32_16x16x4_f32(
                false, Afrag, false, Bfrag, (short)0, C, false, false);
        #pragma unroll
        for (int v = 0; v < 8; ++v)
            sR1[w][m0 + v + 8 * half][lm] = C[v];
    }
    __syncthreads();

    // ---- D = G^T(32x48) x R1(48x16): 2 m-tiles x 12 k-steps ----
    #pragma unroll
    for (int mt = 0; mt < 2; ++mt) {
        const int e0 = mt * 16;
        v8f acc = {};
        #pragma unroll
        for (int kk = 0; kk < 12; ++kk) {
            const int k4 = kk * 4;
            v2f Af, Bf;
            // A[e][k] = G^T[e0+e][k4+K0+k] = G[k4+K0+k][e0+e]
            Af[0] = sG[w][k4 + K0 + 0][e0 + lm];
            Af[1] = sG[w][k4 + K0 + 1][e0 + lm];
            Bf[0] = sR1[w][k4 + K0 + 0][lm];
            Bf[1] = sR1[w][k4 + K0 + 1][lm];
            acc = __builtin_amdgcn_wmma_f32_16x16x4_f32(
                      false, Af, false, Bf, (short)0, acc, false, false);
        }
        float* outp = out + (size_t)bp * (EMBD * SUBD);
        #pragma unroll
        for (int v = 0; v < 8; ++v)
            outp[(e0 + v + 8 * half) * SUBD + lm] = acc[v];
    }
}

extern "C" void kernel_launch(void* const* d_in, const int* in_sizes, int n_in,
                              void* d_out, int out_size, void* d_ws, size_t ws_size,
                              hipStream_t stream) {
    const float* pos   = (const float*)d_in[0];
    const int*   types = (const int*)d_in[1];
    const int*   neigh = (const int*)d_in[2];
    float*       wbuf  = (float*)d_ws;

    // Pack the 30 weight/bias arrays (pairs "00","01","11" x 5 layers) into
    // one contiguous scratch block matching the kernel's per-pair layout.
    static const int DIMS[6] = {1, 2, 4, 8, 16, 32};
    size_t off = 0;
    int argi = 3;
    for (int pair = 0; pair < 3; ++pair) {
        for (int l = 0; l < 5; ++l) {
            const size_t wsz = (size_t)DIMS[l] * DIMS[l + 1];
            const size_t bsz = (size_t)DIMS[l + 1];
            hipMemcpyAsync(wbuf + off, d_in[argi++], wsz * sizeof(float),
                           hipMemcpyDeviceToDevice, stream);
            off += wsz;
            hipMemcpyAsync(wbuf + off, d_in[argi++], bsz * sizeof(float),
                           hipMemcpyDeviceToDevice, stream);
            off += bsz;
        }
    }

    dim3 grid((NSNAP * NPTS) / 4);   // 8192 blocks, 4 points (waves) each
    dim3 block(128);
    descriptor_wmma_kernel<<<grid, block, 0, stream>>>(pos, types, neigh, wbuf,
                                                       (float*)d_out);
}